// Model_12661563588828
// MI455X (gfx1250) — compile-verified
//
#include <hip/hip_runtime.h>
#include <stdint.h>

// ---------------------------------------------------------------------------
// CDNA5 feature gates (host pass must not see amdgcn builtins)
// ---------------------------------------------------------------------------
#if defined(__HIP_DEVICE_COMPILE__) && defined(__gfx1250__) && \
    __has_builtin(__builtin_amdgcn_tensor_load_to_lds) &&      \
    __has_builtin(__builtin_amdgcn_s_wait_tensorcnt)
#define ATHENA_TDM 1
#else
#define ATHENA_TDM 0
#endif

#if defined(__HIP_DEVICE_COMPILE__) && defined(__gfx1250__) && \
    __has_builtin(__builtin_amdgcn_wmma_f32_16x16x4_f32)
#define ATHENA_WMMA 1
#else
#define ATHENA_WMMA 0
#endif

// Compile-time proof that the device pass really takes the CDNA5 paths:
// if either gate is off for gfx1250, fail the build instead of silently
// falling back (compile success == TDM + WMMA instructions emitted).
#if defined(__HIP_DEVICE_COMPILE__) && defined(__gfx1250__)
#if !ATHENA_TDM
#error "gfx1250 device pass lost __builtin_amdgcn_tensor_load_to_lds"
#endif
#if !ATHENA_WMMA
#error "gfx1250 device pass lost __builtin_amdgcn_wmma_f32_16x16x4_f32"
#endif
#endif

typedef float v2f __attribute__((ext_vector_type(2)));
typedef float v8f __attribute__((ext_vector_type(8)));
typedef unsigned int v4u __attribute__((ext_vector_type(4)));
typedef int v4i __attribute__((ext_vector_type(4)));
typedef int v8i __attribute__((ext_vector_type(8)));

static constexpr int IMG    = 1024;
static constexpr int TILE   = 64;
static constexpr int STEP   = 24;
static constexpr int NT     = 41;        // (1024-64)/24+1
static constexpr int ND     = 13;        // displacements per axis
static constexpr int NDISP  = ND * ND;   // 169
static constexpr int KPAD   = 44;        // 41 padded to multiple of 4
static constexpr float MIN_OVERLAP = 500.0f;
static constexpr float SMOOTH = 0.01f;
static constexpr float LR     = 0.1f;

// ---------------------------------------------------------------------------
// K1: block match.  One workgroup (8 waves) per tile.  Src tile staged into
// LDS via the Tensor Data Mover; each wave owns a strided subset of the 169
// displacement candidates; lanes split pixels, wave-reduce with shfl_xor.
// ---------------------------------------------------------------------------
__global__ __launch_bounds__(256) void bm_kernel(const float* __restrict__ src,
                                                 const float* __restrict__ tgt,
                                                 float* __restrict__ G) {
  __shared__ float s_src[TILE * TILE];
  __shared__ float s_cost[8];
  __shared__ int   s_di[8];

  const int tx = blockIdx.x, ty = blockIdx.y;
  const int xs = tx * STEP, ys = ty * STEP;
  const int tid  = threadIdx.x;
  const int wave = tid >> 5, lane = tid & 31;

#if ATHENA_TDM
  if (wave == 0) {
    // Tensor DMA descriptor: 2D tile 64x64 f32 from the 1024x1024 src image.
    unsigned lds_base = (unsigned)(uintptr_t)(&s_src[0]);
    unsigned long long ga =
        (unsigned long long)(uintptr_t)src +
        4ull * ((unsigned long long)ys * (unsigned long long)IMG +
                (unsigned long long)xs);
    v4u g0;
    g0[0] = 1u;                                     // count=1, user descriptor
    g0[1] = lds_base;                               // lds_addr (bytes)
    g0[2] = (unsigned)(ga & 0xFFFFFFFFull);         // global_addr[31:0]
    g0[3] = (unsigned)((ga >> 32) & 0x1FFFFFFull)   // global_addr[56:32]
            | 0x80000000u;                          // type=2 ("image")
    v8i g1;
    g1[0] = 0x20000;                 // data_size=2 (4 bytes), no multicast/pad
    g1[1] = (int)(1024u << 16);      // tensor_dim0[15:0] @ bits[63:48]
    g1[2] = (int)(1024u << 16);      // tensor_dim1[15:0] @ bits[95:80]
    g1[3] = (int)(64u << 16);        // tile_dim0 @ bits[127:112]
    g1[4] = 64;                      // tile_dim1 @ bits[143:128]; tile_dim2=0
    g1[5] = 1024;                    // tensor_dim0_stride[31:0]
    g1[6] = 0;
    g1[7] = 0;
    v4i gz = {0, 0, 0, 0};
#if __clang_major__ >= 23
    v8i gz8 = {0, 0, 0, 0, 0, 0, 0, 0};
    __builtin_amdgcn_tensor_load_to_lds(g0, g1, gz, gz, gz8, 0);
#else
    __builtin_amdgcn_tensor_load_to_lds(g0, g1, gz, gz, 0);
#endif
    __builtin_amdgcn_s_wait_tensorcnt(0);
  }
#else
  for (int i = tid; i < TILE * TILE; i += 256)
    s_src[i] = src[(ys + (i >> 6)) * IMG + xs + (i & 63)];
#endif
  __syncthreads();

  float bestC = __builtin_inff();
  int bestDi = 0;

  for (int di = wave; di < NDISP; di += 8) {
    const int dy = -48 + 8 * (di / ND);
    const int dx = -48 + 8 * (di % ND);
    const int y0 = max(0, -(ys + dy));
    const int y1 = min(TILE, IMG - (ys + dy));
    const int x0 = max(0, -(xs + dx));
    const int x1 = min(TILE, IMG - (xs + dx));
    const int hh = y1 - y0, ww = x1 - x0;
    const float cnt = (hh > 0 && ww > 0) ? (float)(hh * ww) : 0.0f;

    float ssd = 0.0f;
    for (int y = y0; y < y1; ++y) {
      const float* trow = tgt + (ys + y + dy) * IMG + xs + dx;
      const float* srow = s_src + y * TILE;
      for (int x = x0 + lane; x < x1; x += 32) {
        const float d = srow[x] - trow[x];
        ssd = fmaf(d, d, ssd);
      }
    }
    for (int m = 16; m > 0; m >>= 1) ssd += __shfl_xor(ssd, m, 32);

    const float cost =
        (cnt >= MIN_OVERLAP) ? ssd / fmaxf(cnt, 1.0f) : __builtin_inff();
    if (cost < bestC) { bestC = cost; bestDi = di; }  // earliest di wins ties
  }

  if (lane == 0) { s_cost[wave] = bestC; s_di[wave] = bestDi; }
  __syncthreads();

  if (tid == 0) {
    float bc = s_cost[0];
    int bd = s_di[0];
    for (int w2 = 1; w2 < 8; ++w2) {
      const float c2 = s_cost[w2];
      const int d2 = s_di[w2];
      if (c2 < bc || (c2 == bc && d2 < bd)) { bc = c2; bd = d2; }
    }
    const int dy = -48 + 8 * (bd / ND);
    const int dx = -48 + 8 * (bd % ND);
    const int t = ty * NT + tx;
    G[t] = -(float)dy;                // channel 0 (row displacement)
    G[NT * NT + t] = -(float)dx;      // channel 1 (col displacement)
  }
}

// ---------------------------------------------------------------------------
// K2: resize weight matrix Wm[1024][44] (jax.image.resize bilinear weights:
// half-pixel centers, edge renormalization == clamp).  Same for both axes.
// ---------------------------------------------------------------------------
__global__ void weights_kernel(float* __restrict__ Wm) {
  const int i = blockIdx.x * blockDim.x + threadIdx.x;
  if (i >= IMG) return;
  const float s = (i + 0.5f) * ((float)NT / (float)IMG) - 0.5f;
  const float fl = floorf(s);
  const int a0 = (int)fl;
  const float f = s - fl;
  float* row = Wm + i * KPAD;
  for (int a = 0; a < KPAD; ++a) row[a] = 0.0f;
  if (a0 < 0)            row[0] = 1.0f;
  else if (a0 >= NT - 1) row[NT - 1] = 1.0f;
  else { row[a0] = 1.0f - f; row[a0 + 1] = f; }
}

// ---------------------------------------------------------------------------
// K3: T[c][a][j] = sum_b G[c][a][b] * Wx[j][b]  (2 nonzero taps per j)
// Rows a in [41,44) are zero so the dense K=44 WMMA sum stays exact.
// ---------------------------------------------------------------------------
__global__ void t_kernel(const float* __restrict__ G, float* __restrict__ T) {
  const int idx = blockIdx.x * 256 + threadIdx.x;
  if (idx >= 2 * KPAD * IMG) return;
  const int j = idx & (IMG - 1);
  const int a = (idx >> 10) % KPAD;
  const int c = idx / (KPAD * IMG);
  float v = 0.0f;
  if (a < NT) {
    const float s = (j + 0.5f) * ((float)NT / (float)IMG) - 0.5f;
    const float fl = floorf(s);
    const int b0 = (int)fl;
    const float f = s - fl;
    const float* g = G + c * NT * NT + a * NT;
    if (b0 < 0)            v = g[0];
    else if (b0 >= NT - 1) v = g[NT - 1];
    else                   v = (1.0f - f) * g[b0] + f * g[b0 + 1];
  }
  T[c * KPAD * IMG + a * IMG + j] = v;
}

// ---------------------------------------------------------------------------
// K4: O = Wy(1024x44) x T(44x1024) per channel via V_WMMA_F32_16X16X4_F32.
// One wave per 16x16 output tile; 11 k-steps of K=4.
// A layout (ISA 7.12.2): lanes 0-15 M=0..15 {v0=K0,v1=K1}; lanes 16-31 {K2,K3}.
// B layout: lanes 0-15 N=0..15 {v0=K0,v1=K1}; lanes 16-31 {K2,K3}.
// C/D: VGPR r -> M=r (lanes 0-15) / M=r+8 (lanes 16-31), N=lane%16.
// ---------------------------------------------------------------------------
__global__ __launch_bounds__(32) void resize_wmma_kernel(
    const float* __restrict__ Wm, const float* __restrict__ T,
    float* __restrict__ res) {
  const int jt = blockIdx.x, it = blockIdx.y, c = blockIdx.z;
  const int lane = threadIdx.x;
  const int half = lane >> 4;
  const int l = lane & 15;
  const int i0 = it * 16, j0 = jt * 16;
  const float* Trow = T + c * KPAD * IMG;

#if ATHENA_WMMA
  v8f acc = {0, 0, 0, 0, 0, 0, 0, 0};
  for (int k = 0; k < KPAD; k += 4) {
    const int ka = k + half * 2;
    v2f a, b;
    a.x = Wm[(i0 + l) * KPAD + ka];
    a.y = Wm[(i0 + l) * KPAD + ka + 1];
    b.x = Trow[ka * IMG + j0 + l];
    b.y = Trow[(ka + 1) * IMG + j0 + l];
    acc = __builtin_amdgcn_wmma_f32_16x16x4_f32(
        /*neg_a=*/false, a, /*neg_b=*/false, b,
        /*c_mod=*/(short)0, acc, /*reuse_a=*/false, /*reuse_b=*/false);
  }
  for (int r = 0; r < 8; ++r) {
    const int i = i0 + r + half * 8;
    const int j = j0 + l;
    res[(i * IMG + j) * 2 + c] = acc[r];
  }
#else
  for (int r = 0; r < 8; ++r) {
    const int i = i0 + r + half * 8;
    const int j = j0 + l;
    float acc2 = 0.0f;
    for (int a2 = 0; a2 < KPAD; ++a2)
      acc2 = fmaf(Wm[i * KPAD + a2], Trow[a2 * IMG + j], acc2);
    res[(i * IMG + j) * 2 + c] = acc2;
  }
#endif
}

// ---------------------------------------------------------------------------
// K5 (x100): fused gradient + update, double-buffered.
// data grad:  2*(warp - tgt) * d(bilinear)/d(res)   (clip-aware, wy unclipped)
// smooth grad: 2*sum of one-sided forward differences, SMOOTH-weighted.
// ---------------------------------------------------------------------------
__global__ __launch_bounds__(256) void opt_kernel(const float* __restrict__ src,
                                                  const float* __restrict__ tgt,
                                                  const float* __restrict__ rin,
                                                  float* __restrict__ rout) {
  const int p = blockIdx.x * 256 + threadIdx.x;
  if (p >= IMG * IMG) return;
  const int y = p >> 10, x = p & (IMG - 1);

  const float r0 = rin[2 * p];
  const float r1 = rin[2 * p + 1];
  const float sy = (float)y + r0;
  const float sx = (float)x + r1;
  const float fy = floorf(sy), fx = floorf(sx);
  const float wy = sy - fy, wx = sx - fx;
  int y0 = (int)fy; y0 = min(max(y0, 0), IMG - 1);
  const int y1 = min(y0 + 1, IMG - 1);
  int x0 = (int)fx; x0 = min(max(x0, 0), IMG - 1);
  const int x1 = min(x0 + 1, IMG - 1);

  const float v00 = src[y0 * IMG + x0];
  const float v01 = src[y0 * IMG + x1];
  const float v10 = src[y1 * IMG + x0];
  const float v11 = src[y1 * IMG + x1];

  const float w = v00 * (1.0f - wy) * (1.0f - wx) + v01 * (1.0f - wy) * wx +
                  v10 * wy * (1.0f - wx) + v11 * wy * wx;
  const float diff = w - tgt[p];
  const float dwdy = (v10 - v00) * (1.0f - wx) + (v11 - v01) * wx;
  const float dwdx = (v01 - v00) * (1.0f - wy) + (v11 - v10) * wy;

  float g0 = 2.0f * diff * dwdy;
  float g1 = 2.0f * diff * dwdx;

#pragma unroll
  for (int c = 0; c < 2; ++c) {
    const float rc = (c == 0) ? r0 : r1;
    float acc = 0.0f;
    if (y > 0)       acc += rc - rin[2 * ((y - 1) * IMG + x) + c];
    if (y < IMG - 1) acc -= rin[2 * ((y + 1) * IMG + x) + c] - rc;
    if (x > 0)       acc += rc - rin[2 * (y * IMG + x - 1) + c];
    if (x < IMG - 1) acc -= rin[2 * (y * IMG + x + 1) + c] - rc;
    const float gs = SMOOTH * 2.0f * acc;
    if (c == 0) g0 += gs; else g1 += gs;
  }

  rout[2 * p]     = r0 - LR * g0;
  rout[2 * p + 1] = r1 - LR * g1;
}

// ---------------------------------------------------------------------------
// K6: final scale by 2/H
// ---------------------------------------------------------------------------
__global__ void finish_kernel(const float* __restrict__ r,
                              float* __restrict__ out) {
  const int i = blockIdx.x * 256 + threadIdx.x;
  if (i < 2 * IMG * IMG) out[i] = r[i] * (2.0f / (float)IMG);
}

// ---------------------------------------------------------------------------
extern "C" void kernel_launch(void* const* d_in, const int* in_sizes, int n_in,
                              void* d_out, int out_size, void* d_ws,
                              size_t ws_size, hipStream_t stream) {
  (void)in_sizes; (void)n_in; (void)out_size; (void)ws_size;
  const float* src = (const float*)d_in[0];
  const float* tgt = (const float*)d_in[1];
  float* wsf = (float*)d_ws;

  float* G  = wsf;                    // 2*41*41      (pad to 4096)
  float* Wm = wsf + 4096;             // 1024*44 = 45056
  float* T  = wsf + 65536;            // 2*44*1024 = 90112
  float* rA = wsf + 262144;           // 2*1024*1024
  float* rB = rA + 2 * IMG * IMG;     // 2*1024*1024  (total ~17.8 MB)

  bm_kernel<<<dim3(NT, NT), 256, 0, stream>>>(src, tgt, G);
  weights_kernel<<<(IMG + 255) / 256, 256, 0, stream>>>(Wm);
  t_kernel<<<(2 * KPAD * IMG + 255) / 256, 256, 0, stream>>>(G, T);
  resize_wmma_kernel<<<dim3(IMG / 16, IMG / 16, 2), 32, 0, stream>>>(Wm, T, rA);

  for (int it = 0; it < 100; ++it) {
    opt_kernel<<<(IMG * IMG + 255) / 256, 256, 0, stream>>>(src, tgt, rA, rB);
    float* tmp = rA; rA = rB; rB = tmp;
  }

  finish_kernel<<<(2 * IMG * IMG + 255) / 256, 256, 0, stream>>>(
      rA, (float*)d_out);
}